// Entmax_33277406609981
// MI455X (gfx1250) — compile-verified
//
#include <hip/hip_runtime.h>
#include <cstdint>

#define COLS     32000
#define THREADS  1024
#define CHUNKS   8          // float4 chunks per thread: 8*4 = 32 elems
#define ACTIVE   1000       // 1000 threads * 32 elems = 32000
#define NWAVES   (THREADS / 32)
#define N_ITER   50

__device__ __forceinline__ float wave_sum(float v) {
#pragma unroll
  for (int off = 16; off > 0; off >>= 1) v += __shfl_xor(v, off, 32);
  return v;
}
__device__ __forceinline__ float wave_max(float v) {
#pragma unroll
  for (int off = 16; off > 0; off >>= 1) v = fmaxf(v, __shfl_xor(v, off, 32));
  return v;
}
__device__ __forceinline__ float wave_min(float v) {
#pragma unroll
  for (int off = 16; off > 0; off >>= 1) v = fminf(v, __shfl_xor(v, off, 32));
  return v;
}

// Issue per-lane async copies of this thread's 128-byte slice of row `base`
// into the LDS row buffer. Tracked by ASYNCcnt; consumes no data VGPRs.
__device__ __forceinline__ void async_row_to_lds(const float* base,
                                                 unsigned lds_base,
                                                 int tid) {
#pragma unroll
  for (int k = 0; k < CHUNKS; ++k) {
    unsigned off = (unsigned)(tid * (CHUNKS * 16) + k * 16);
    unsigned la  = lds_base + off;
    asm volatile("global_load_async_to_lds_b128 %0, %1, %2"
                 :: "v"(la), "v"(off), "s"(base)
                 : "memory");
  }
}

__global__ __launch_bounds__(THREADS, 1)
void entmax_bisect_kernel(const float* __restrict__ x,
                          float* __restrict__ out,
                          int rows) {
  extern __shared__ float4 lbuf[];        // 8000 float4 = 128000 bytes
  __shared__ float red[NWAVES];

  const int tid    = (int)threadIdx.x;
  const int lane   = tid & 31;
  const int wv     = tid >> 5;
  const bool active = (tid < ACTIVE);
  const unsigned lds_base = (unsigned)(uintptr_t)lbuf;  // low 32 bits = LDS byte offset

  // Prefetch first row for this persistent block.
  int row = (int)blockIdx.x;
  if (row < rows && active) {
    async_row_to_lds(x + (size_t)row * COLS, lds_base, tid);
  }

  for (; row < rows; row += (int)gridDim.x) {
    // Wait for this wave's async copies; barrier makes all waves' data visible.
    asm volatile("s_wait_asynccnt 0" ::: "memory");
    __syncthreads();

    // LDS -> registers (ds_load_b128 x8). Inactive tail threads get -inf,
    // which contributes 0 after clip().
    float4 zr[CHUNKS];
    if (active) {
#pragma unroll
      for (int k = 0; k < CHUNKS; ++k) zr[k] = lbuf[tid * CHUNKS + k];
    } else {
#pragma unroll
      for (int k = 0; k < CHUNKS; ++k)
        zr[k] = make_float4(-INFINITY, -INFINITY, -INFINITY, -INFINITY);
    }
    // Ensure all ds_loads have drained before anyone overwrites the buffer.
    asm volatile("s_wait_dscnt 0" ::: "memory");
    __syncthreads();

    // Kick async prefetch of the next row; overlaps with the bisection loop.
    {
      int nrow = row + (int)gridDim.x;
      if (nrow < rows && active) {
        async_row_to_lds(x + (size_t)nrow * COLS, lds_base, tid);
      }
    }

    // ---- Row max / min (block reduction) ----
    float mx = -INFINITY, mn = INFINITY;
#pragma unroll
    for (int k = 0; k < CHUNKS; ++k) {
      mx = fmaxf(mx, fmaxf(fmaxf(zr[k].x, zr[k].y), fmaxf(zr[k].z, zr[k].w)));
    }
    if (active) {
#pragma unroll
      for (int k = 0; k < CHUNKS; ++k) {
        mn = fminf(mn, fminf(fminf(zr[k].x, zr[k].y), fminf(zr[k].z, zr[k].w)));
      }
    }
    mx = wave_max(mx);
    if (lane == 0) red[wv] = mx;
    __syncthreads();
    mx = wave_max(red[lane]);
    __syncthreads();
    mn = wave_min(mn);
    if (lane == 0) red[wv] = mn;
    __syncthreads();
    mn = wave_min(red[lane]);
    __syncthreads();

    // z = (ALPHA-1) * (x - max) = 0.5 * x_shifted ; tau bounds on x_shifted scale.
    const float half = 0.5f;
#pragma unroll
    for (int k = 0; k < CHUNKS; ++k) {
      zr[k].x = (zr[k].x - mx) * half;
      zr[k].y = (zr[k].y - mx) * half;
      zr[k].z = (zr[k].z - mx) * half;
      zr[k].w = (zr[k].w - mx) * half;
    }
    float tmin = (mn - mx) - 1.0f;
    float tmax = 0.0f;

    // ---- 50 bisection iterations, fully register-resident ----
    for (int it = 0; it < N_ITER; ++it) {
      float tau = 0.5f * (tmin + tmax);
      float p = 0.0f;
#pragma unroll
      for (int k = 0; k < CHUNKS; ++k) {
        float a;
        a = fmaxf(zr[k].x - tau, 0.0f); p = fmaf(a, a, p);
        a = fmaxf(zr[k].y - tau, 0.0f); p = fmaf(a, a, p);
        a = fmaxf(zr[k].z - tau, 0.0f); p = fmaf(a, a, p);
        a = fmaxf(zr[k].w - tau, 0.0f); p = fmaf(a, a, p);
      }
      p = wave_sum(p);
      if (lane == 0) red[wv] = p;
      __syncthreads();
      float s = wave_sum(red[lane]);   // broadcast total to every thread
      __syncthreads();
      float c = s - 1.0f;
      if (c < 0.0f) tmin = tau;        // constraint<0 -> raise lower bound
      if (c > 0.0f) tmax = tau;        // constraint>0 -> lower upper bound
    }

    // ---- Final y, normalize, store ----
    float tau = 0.5f * (tmin + tmax);
    float p = 0.0f;
#pragma unroll
    for (int k = 0; k < CHUNKS; ++k) {
      float a;
      a = fmaxf(zr[k].x - tau, 0.0f); zr[k].x = a * a; p += zr[k].x;
      a = fmaxf(zr[k].y - tau, 0.0f); zr[k].y = a * a; p += zr[k].y;
      a = fmaxf(zr[k].z - tau, 0.0f); zr[k].z = a * a; p += zr[k].z;
      a = fmaxf(zr[k].w - tau, 0.0f); zr[k].w = a * a; p += zr[k].w;
    }
    p = wave_sum(p);
    if (lane == 0) red[wv] = p;
    __syncthreads();
    float s = wave_sum(red[lane]);
    __syncthreads();
    float inv = 1.0f / (s + 1e-12f);

    if (active) {
      float4* orow = (float4*)(out + (size_t)row * COLS);
#pragma unroll
      for (int k = 0; k < CHUNKS; ++k) {
        float4 o = zr[k];
        o.x *= inv; o.y *= inv; o.z *= inv; o.w *= inv;
        orow[tid * CHUNKS + k] = o;     // global_store_b128
      }
    }
    // Loop back: wait on the prefetch we issued above.
  }
}

extern "C" void kernel_launch(void* const* d_in, const int* in_sizes, int n_in,
                              void* d_out, int out_size, void* d_ws, size_t ws_size,
                              hipStream_t stream) {
  const float* x = (const float*)d_in[0];
  float* out     = (float*)d_out;
  int rows       = in_sizes[0] / COLS;

  const size_t shbytes = (size_t)COLS * sizeof(float);   // 128000 B row buffer
  (void)hipFuncSetAttribute((const void*)entmax_bisect_kernel,
                            hipFuncAttributeMaxDynamicSharedMemorySize,
                            (int)(160 * 1024));

  int grid = rows < 512 ? rows : 512;   // persistent blocks, ~8 rows each
  entmax_bisect_kernel<<<grid, THREADS, shbytes, stream>>>(x, out, rows);
}